// EncoderRNN_69114613730467
// MI455X (gfx1250) — compile-verified
//
#include <hip/hip_runtime.h>
#include <hip/hip_bf16.h>

#define TSEQ 4096
#define EMB  512
#define HID  512
#define N3   1536          // 3*HID
#define NBLK 16            // cluster size (max), one WGP each
#define RPB  96            // W_hh rows per block (3 gates * 32 hidden)
#define LSTR 520           // LDS row stride (floats): 512 + TDM pad (4 DW per 256 DW)

typedef float    v2f __attribute__((ext_vector_type(2)));
typedef float    v8f __attribute__((ext_vector_type(8)));
typedef unsigned v4u __attribute__((ext_vector_type(4)));
typedef int      v4i __attribute__((ext_vector_type(4)));
typedef int      v8i __attribute__((ext_vector_type(8)));

// ---------------- TDM: tensor_load_to_lds wrapper (toolchain-arity hedge) -----------------
__device__ __forceinline__ void tdm_load(v4u g0, v8i g1, v4i g2, v4i g3) {
#if defined(__clang_major__) && (__clang_major__ >= 23)
    v8i z = {0, 0, 0, 0, 0, 0, 0, 0};
    __builtin_amdgcn_tensor_load_to_lds(g0, g1, g2, g3, z, 0);
#else
    __builtin_amdgcn_tensor_load_to_lds(g0, g1, g2, g3, 0);
#endif
}

// Build D# group0: count=1, optional gather mode, LDS byte addr, 57b global addr, type=2.
__device__ __forceinline__ v4u tdm_g0(unsigned lds_bytes, const void* gptr, bool gather) {
    unsigned long long ga = (unsigned long long)(uintptr_t)gptr;
    v4u g0;
    g0[0] = 1u | (gather ? (1u << 31) : 0u);        // count=1; gather_index_size=16-bit
    g0[1] = lds_bytes;                              // lds_addr
    g0[2] = (unsigned)(ga & 0xFFFFFFFFull);         // global_addr[31:0]
    g0[3] = (unsigned)((ga >> 32) & 0x01FFFFFFull)  // global_addr[56:32]
          | (2u << 30);                             // type = 2 ("image")
    return g0;
}

// Build D# group1: f32 elements, 2D tile, pad 4 DWORDs every 256 DWORDs (row stride 520).
__device__ __forceinline__ v8i tdm_g1(unsigned tdim0, unsigned tdim1,
                                      unsigned tile0, unsigned tile1) {
    v8i g1;
    g1[0] = (int)((2u << 16)      // data_size = 4B  [17:16]
          |       (1u << 20)      // pad_enable      [20]
          |       (7u << 22)      // pad_interval: 256 DWORDs [24:22]
          |       (3u << 25));    // pad_amount: 4 DWORDs     [31:25]
    g1[1] = (int)((tdim0 & 0xFFFFu) << 16);                         // tensor_dim0[15:0]
    g1[2] = (int)(((tdim0 >> 16) & 0xFFFFu) | ((tdim1 & 0xFFFFu) << 16));
    g1[3] = (int)(((tdim1 >> 16) & 0xFFFFu) | ((tile0 & 0xFFFFu) << 16)); // tile_dim0
    g1[4] = (int)(tile1 & 0xFFFFu);                                 // tile_dim1 (gather: #idx)
    g1[5] = (int)tdim0;                                             // tensor_dim0_stride[31:0]
    g1[6] = 0;
    g1[7] = 0;
    return g1;
}

// ---------------- barrier counter init (re-zeroed every launch: deterministic) ------------
__global__ void init_cnt(unsigned* cnt) {
    if (threadIdx.x == 0 && blockIdx.x == 0) *cnt = 0u;
}

// ---------------- Phase 1: xproj = gather(emb, ids) @ W_ih^T + b_ih  (f32 WMMA) ----------
// grid = (4096/16, 1536/128), block = 256 (8 waves); each wave owns one 16x16 tile.
// X tile (16 gathered rows of emb) is fetched by one TDM gather-mode DMA into LDS.
__global__ void xproj_wmma(const int* __restrict__ ids,
                           const float* __restrict__ emb,
                           const float* __restrict__ Wih,
                           const float* __restrict__ bih,
                           float* __restrict__ xp) {
    __shared__ float Xs[16 * LSTR];
    const int tid = threadIdx.x;
    const int mt  = blockIdx.x;                    // 16-row tile of tokens

    if (tid == 0) {
        // Pack 16 token ids (< 32768 fits i16) as TDM gather indices.
        v4i g2, g3;
#pragma unroll
        for (int i = 0; i < 4; ++i) {
            unsigned lo = (unsigned)ids[mt * 16 + 2 * i]     & 0xFFFFu;
            unsigned hi = (unsigned)ids[mt * 16 + 2 * i + 1] & 0xFFFFu;
            g2[i] = (int)(lo | (hi << 16));
            unsigned lo2 = (unsigned)ids[mt * 16 + 8 + 2 * i]     & 0xFFFFu;
            unsigned hi2 = (unsigned)ids[mt * 16 + 8 + 2 * i + 1] & 0xFFFFu;
            g3[i] = (int)(lo2 | (hi2 << 16));
        }
        v4u g0 = tdm_g0((unsigned)(uintptr_t)Xs, emb, /*gather=*/true);
        v8i g1 = tdm_g1(/*tensor_dim0=*/EMB, /*tensor_dim1(rows)=*/32000,
                        /*tile_dim0=*/EMB, /*tile_dim1=#indices*/16);
        tdm_load(g0, g1, g2, g3);                   // tensor_load_to_lds (gather, padded)
        __builtin_amdgcn_s_wait_tensorcnt(0);
    }
    __syncthreads();

    const int wave = tid >> 5;
    const int lane = tid & 31;
    const int half = lane >> 4;                    // 0: lanes 0-15, 1: lanes 16-31
    const int l15  = lane & 15;
    const int nt   = blockIdx.y * 8 + wave;        // 16-col tile of 3*HID
    const int ncol = nt * 16 + l15;

    const float* wcol = Wih + (size_t)ncol * EMB;  // W_ih row ncol == column of W_ih^T
    const float* xrow = Xs + l15 * LSTR;           // A-matrix row M = l15 (padded layout)

    v8f acc = {};
    for (int k0 = 0; k0 < EMB; k0 += 4) {
        const int kk = k0 + 2 * half;              // K pair per lane half (ISA 16x4 layout)
        const int ka = kk + (kk >= 256 ? 4 : 0);   // TDM pad skip at 256-DWORD boundary
        v2f a, b;
        a[0] = xrow[ka];  a[1] = xrow[ka + 1];     // A: 16x4 f32 fragment
        b[0] = wcol[kk];  b[1] = wcol[kk + 1];     // B: 4x16 f32 fragment
        acc = __builtin_amdgcn_wmma_f32_16x16x4_f32(
                  false, a, false, b, (short)0, acc, false, false);
    }

    const float bias = bih[ncol];
#pragma unroll
    for (int v = 0; v < 8; ++v) {
        const int row = mt * 16 + v + 8 * half;    // C layout: VGPR v -> M=v / M=v+8
        xp[(size_t)row * N3 + ncol] = acc[v] + bias;
    }
}

// ---------------- cross-WGP barrier: atomic counter + cluster HW barrier ------------------
__device__ __forceinline__ void gbar(unsigned* cnt, unsigned epoch) {
    __syncthreads();
    if (threadIdx.x == 0) {
        __threadfence();                                        // make h/out stores visible
        __hip_atomic_fetch_add(cnt, 1u, __ATOMIC_RELEASE, __HIP_MEMORY_SCOPE_AGENT);
        const unsigned target = epoch * NBLK;
        while (__hip_atomic_load(cnt, __ATOMIC_ACQUIRE, __HIP_MEMORY_SCOPE_AGENT) < target)
            __builtin_amdgcn_s_sleep(1);
        // Fast HW path when dispatched as a 16-WG cluster (s_barrier_signal/-wait -3);
        // architecturally NOP when ClusterID==0.
        __builtin_amdgcn_s_cluster_barrier();
    }
    __syncthreads();
}

// ---------------- Phase 2: sequential GRU scan, W_hh sharded into 16 WGPs' LDS ------------
// Block b owns hidden slice [32b, 32b+32): W_hh rows {j, 512+j, 1024+j}, staged by 3 TDM DMAs.
__global__ void gru_scan(const float* __restrict__ xp,
                         const float* __restrict__ h0,
                         const float* __restrict__ Whh,
                         const float* __restrict__ bhh,
                         float* __restrict__ out,
                         float* hbuf,               // [2][HID] ping-pong state in L2
                         unsigned* cnt) {
    extern __shared__ float smem[];
    float* Wsh = smem;                  // RPB * LSTR  (~195 KB, fits 320 KB WGP LDS)
    float* hsh = Wsh + RPB * LSTR;      // HID
    float* gsh = hsh + HID;             // RPB
    float* bsl = gsh + RPB;             // RPB

    const int tid = threadIdx.x;
    const int j0  = blockIdx.x * 32;

    // Kick off 3 TDM loads: 32x512 f32 tile per gate, DMA'd into padded LDS rows.
    if (tid == 0) {
        v8i g1 = tdm_g1(/*tensor_dim0=*/HID, /*tensor_dim1=*/32,
                        /*tile_dim0=*/HID, /*tile_dim1=*/32);
        v4i gz = {0, 0, 0, 0};
#pragma unroll
        for (int g = 0; g < 3; ++g) {
            const float* src = Whh + (size_t)(g * HID + j0) * HID;
            unsigned lds = (unsigned)(uintptr_t)(Wsh + g * 32 * LSTR);
            v4u g0 = tdm_g0(lds, src, /*gather=*/false);
            tdm_load(g0, g1, gz, gz);
        }
    }

    // Overlap DMA with bias staging + h0 publication.
    if (tid < RPB) {
        const int G = (tid >> 5) * HID + j0 + (tid & 31);
        bsl[tid] = bhh[G];
    }
    if (blockIdx.x == 0) {              // publish h0 into ping buffer
        for (int i = tid; i < HID; i += 256)
            __hip_atomic_store(&hbuf[i], h0[i], __ATOMIC_RELAXED, __HIP_MEMORY_SCOPE_AGENT);
    }
    if (tid == 0) __builtin_amdgcn_s_wait_tensorcnt(0);   // W slice resident in LDS

    unsigned epoch = 0;
    gbar(cnt, ++epoch);                 // everyone sees h0 + own LDS staged

    for (int t = 0; t < TSEQ; ++t) {
        float* hin  = hbuf + ((t)     & 1) * HID;
        float* hout = hbuf + ((t + 1) & 1) * HID;

        // Fresh agent-scope copy of h_t into LDS (bypasses WGP$).
        for (int i = tid; i < HID; i += 256)
            hsh[i] = __hip_atomic_load(&hin[i], __ATOMIC_RELAXED, __HIP_MEMORY_SCOPE_AGENT);
        __syncthreads();

        // 96 lanes: one full 512-FMA dot each, ds_load_b128 from padded LDS rows.
        if (tid < RPB) {
            const float4* wr0 = (const float4*)(Wsh + tid * LSTR);        // k in [0,256)
            const float4* wr1 = (const float4*)(Wsh + tid * LSTR + 260);  // k in [256,512)
            const float4* hv  = (const float4*)hsh;
            float s = bsl[tid];
#pragma unroll 4
            for (int k4 = 0; k4 < 64; ++k4) {
                const float4 w = wr0[k4], h = hv[k4];
                s += w.x * h.x + w.y * h.y + w.z * h.z + w.w * h.w;
            }
#pragma unroll 4
            for (int k4 = 0; k4 < 64; ++k4) {
                const float4 w = wr1[k4], h = hv[64 + k4];
                s += w.x * h.x + w.y * h.y + w.z * h.z + w.w * h.w;
            }
            gsh[tid] = s;
        }
        __syncthreads();

        // 32 lanes: gates + state update for this block's hidden slice.
        if (tid < 32) {
            const int j = j0 + tid;
            const float* xrow = xp + (size_t)t * N3;
            const float xr = xrow[j], xz = xrow[HID + j], xn = xrow[2 * HID + j];
            if (t + 1 < TSEQ)
                __builtin_prefetch(xp + (size_t)(t + 1) * N3 + j, 0, 1);  // global_prefetch_b8
            const float r = 1.f / (1.f + __expf(-(xr + gsh[tid])));
            const float z = 1.f / (1.f + __expf(-(xz + gsh[32 + tid])));
            const float n = tanhf(xn + r * gsh[64 + tid]);
            const float hnew = (1.f - z) * n + z * hsh[j];
            out[(size_t)t * HID + j] = hnew;
            __hip_atomic_store(&hout[j], hnew, __ATOMIC_RELAXED, __HIP_MEMORY_SCOPE_AGENT);
        }
        gbar(cnt, ++epoch);             // h_{t+1} published device-wide
    }
}

// ------------------------------------------------------------------------------------------
extern "C" void kernel_launch(void* const* d_in, const int* in_sizes, int n_in,
                              void* d_out, int out_size, void* d_ws, size_t ws_size,
                              hipStream_t stream) {
    (void)in_sizes; (void)n_in; (void)out_size; (void)ws_size;
    const int*   ids = (const int*)  d_in[0];
    const float* h0  = (const float*)d_in[1];
    const float* emb = (const float*)d_in[2];
    const float* Wih = (const float*)d_in[3];
    const float* Whh = (const float*)d_in[4];
    const float* bih = (const float*)d_in[5];
    const float* bhh = (const float*)d_in[6];
    float* out = (float*)d_out;

    float*    xp   = (float*)d_ws;                   // 4096*1536 f32 = 24 MB
    float*    hbuf = xp + (size_t)TSEQ * N3;         // 2*512 f32 ping-pong state
    unsigned* cnt  = (unsigned*)(hbuf + 2 * HID);    // barrier counter

    init_cnt<<<1, 1, 0, stream>>>(cnt);

    dim3 g1(TSEQ / 16, N3 / 128);                    // 256 x 12 blocks, 8 waves each
    xproj_wmma<<<g1, 256, 0, stream>>>(ids, emb, Wih, bih, xp);

    const size_t lds = (size_t)(RPB * LSTR + HID + RPB + RPB) * sizeof(float);
    gru_scan<<<NBLK, 256, lds, stream>>>(xp, h0, Whh, bhh, out, hbuf, cnt);
}